// GINLayer_31190052504404
// MI455X (gfx1250) — compile-verified
//
#include <hip/hip_runtime.h>

// ---------------------------------------------------------------------------
// GIN layer for MI455X (gfx1250, wave32).
//   agg = segment_sum(x[src], dst);  h = x + agg
//   out = relu(h@W1 + b1) @ W2 + b2
// Phase 1 (scatter) is L2-atomic/bandwidth bound.  Phase 2 is a fused 2-layer
// MLP on V_WMMA_F32_16X16X4_F32 (exact f32 reference precision).  Weights are
// held in LDS in a K-pair-interleaved layout so every B fragment is a single
// conflict-free ds_load_b64 into an even-aligned VGPR pair (no v_mov fixup).
// Each wave computes 32 rows (two 16-row M tiles) so each B fragment feeds
// two WMMAs.
// ---------------------------------------------------------------------------

#define N_NODES 100000
#define N_EDGES 1600000
#define D       64

typedef __attribute__((ext_vector_type(2))) float v2f;
typedef __attribute__((ext_vector_type(4))) float v4f;
typedef __attribute__((ext_vector_type(8))) float v8f;

// ---------------------------------------------------------------- kernel 1 --
// h0 = x  (accumulator seeded with x: folds the "+x" of GIN into the scatter)
__global__ __launch_bounds__(256) void gin_init(const float* __restrict__ x,
                                                float* __restrict__ h0) {
    size_t i = (size_t)blockIdx.x * blockDim.x + threadIdx.x;
    const size_t n4 = (size_t)N_NODES * D / 4;
    if (i < n4) {
        ((v4f*)h0)[i] = ((const v4f*)x)[i];
    }
}

// ---------------------------------------------------------------- kernel 2 --
// 16 threads per edge; each thread gathers a float4 of x[src] (b128 load) and
// scatters it with 4 global_atomic_add_f32 into h0[dst].  Edge indices are
// the only linearly-streamed data -> prefetch them ahead.
__global__ __launch_bounds__(256) void gin_scatter(const float* __restrict__ x,
                                                   const int* __restrict__ src,
                                                   const int* __restrict__ dst,
                                                   float* __restrict__ h0) {
    size_t t = (size_t)blockIdx.x * blockDim.x + threadIdx.x;
    size_t e = t >> 4;
    int    f = (int)(t & 15) * 4;
    if (e < (size_t)N_EDGES) {
        if (f == 0 && e + 4096 < (size_t)N_EDGES) {
            __builtin_prefetch(src + e + 4096, 0, 0);   // global_prefetch_b8
            __builtin_prefetch(dst + e + 4096, 0, 0);
        }
        int s = src[e];
        int d = dst[e];
        v4f v = *(const v4f*)(x + (size_t)s * D + f);
        float* p = h0 + (size_t)d * D + f;
        atomicAdd(p + 0, v[0]);
        atomicAdd(p + 1, v[1]);
        atomicAdd(p + 2, v[2]);
        atomicAdd(p + 3, v[3]);
    }
}

// ---------------------------------------------------------------- kernel 3 --
// Fused MLP.  128 threads = 4 waves; each wave computes 32 rows (2 M-tiles).
// WMMA fragment layouts per ISA 7.12.2 (32-bit, wave32):
//   A 16x4 : lanes 0-15 hold K=0,1 in v0,v1; lanes 16-31 hold K=2,3.
//   B 4x16 : mirrored, N across lanes.
//   C/D    : VGPR r = row r (lanes 0-15) / row r+8 (lanes 16-31), N = lane&15.
// Swizzled weight LDS layout: (k,col) -> (k>>2)*KSTR + ((k>>1)&1)*LHSTR
//                                        + 2*col + (k&1)
// so a B fragment at (koff, col) is a contiguous b64 at
//   kk*KSTR + lhalf*LHSTR + 2*col.   LHSTR%64 == 32 banks -> lanes 0-15 and
// 16-31 use disjoint bank halves (conflict-free b64 across the wave).
#define MLP_WAVES 4
#define HSTR  68    // padded row stride for the LDS relayout tile
#define KSTR  320   // floats per kk group in swizzled weight layout
#define LHSTR 160   // lhalf offset within a kk group

__global__ __launch_bounds__(128) void gin_mlp(const float* __restrict__ h0,
                                               const float* __restrict__ W1,
                                               const float* __restrict__ b1,
                                               const float* __restrict__ W2,
                                               const float* __restrict__ b2,
                                               float* __restrict__ out) {
    __shared__ float sW1[16 * KSTR];                 // 20 KB
    __shared__ float sW2[16 * KSTR];                 // 20 KB
    __shared__ float sB1[D];
    __shared__ float sB2[D];
    __shared__ float sH[MLP_WAVES][16 * HSTR];       // 17 KB, reused per M-tile

    const int tid = threadIdx.x;

    // Stage weights swizzled: thread loads a float4 of row k, scatters 4 b32.
    for (int i = tid; i < D * D / 4; i += 128) {
        const int k = i >> 4;                        // 0..63
        const int c = (i & 15) << 2;                 // 0..60 step 4
        const int base = (k >> 2) * KSTR + ((k >> 1) & 1) * LHSTR + (k & 1);
        v4f w1 = ((const v4f*)W1)[i];
        v4f w2 = ((const v4f*)W2)[i];
        #pragma unroll
        for (int j = 0; j < 4; ++j) {
            sW1[base + ((c + j) << 1)] = w1[j];
            sW2[base + ((c + j) << 1)] = w2[j];
        }
    }
    if (tid < D) {
        sB1[tid] = b1[tid];
        sB2[tid] = b2[tid];
    }
    __syncthreads();

    const int wave  = tid >> 5;
    const int lane  = tid & 31;
    const int lhalf = lane >> 4;     // 0: K 0,1 / rows 0-7   1: K 2,3 / rows 8-15
    const int l16   = lane & 15;

    const int row0 = (blockIdx.x * MLP_WAVES + wave) * 32;
    // Clamp A-rows so EXEC stays all-ones through every WMMA (tail block).
    const float* a0p = h0 + (size_t)min(row0 + l16,      N_NODES - 1) * D;
    const float* a1p = h0 + (size_t)min(row0 + 16 + l16, N_NODES - 1) * D;

    const v8f vz = {0, 0, 0, 0, 0, 0, 0, 0};
    v8f acc0[4], acc1[4];
    #pragma unroll
    for (int nt = 0; nt < 4; ++nt) { acc0[nt] = vz; acc1[nt] = vz; }

    const float* bw1 = sW1 + lhalf * LHSTR + (l16 << 1);
    const float* bw2 = sW2 + lhalf * LHSTR + (l16 << 1);

    // ---- GEMM1: two 16x64 A tiles vs 64x64 W1; B fragment reused 2x --------
    #pragma unroll
    for (int kk = 0; kk < 16; ++kk) {
        const int koff = kk * 4 + lhalf * 2;
        const v2f a0 = *(const v2f*)(a0p + koff);            // global b64
        const v2f a1 = *(const v2f*)(a1p + koff);
        #pragma unroll
        for (int nt = 0; nt < 4; ++nt) {
            const v2f b = *(const v2f*)(bw1 + kk * KSTR + nt * 32);  // ds b64
            acc0[nt] = __builtin_amdgcn_wmma_f32_16x16x4_f32(
                false, a0, false, b, (short)0, acc0[nt], false, false);
            acc1[nt] = __builtin_amdgcn_wmma_f32_16x16x4_f32(
                false, a1, false, b, (short)0, acc1[nt], false, false);
        }
    }

    // ---- per M-tile: bias+ReLU -> LDS relayout -> GEMM2 -> store -----------
    float* myH = &sH[wave][0];
    const float* aH = myH + l16 * HSTR;

    #pragma unroll
    for (int mt = 0; mt < 2; ++mt) {
        const v8f* acc = (mt == 0) ? acc0 : acc1;

        // bias + ReLU; spill C/D-layout tile to LDS.  (Same-wave DS ops are
        // in-order, so mt=1 stores cannot pass mt=0's A reads below.)
        #pragma unroll
        for (int nt = 0; nt < 4; ++nt) {
            const int   col  = nt * 16 + l16;
            const float bias = sB1[col];
            #pragma unroll
            for (int r = 0; r < 8; ++r) {
                float v = acc[nt][r] + bias;
                v = v > 0.0f ? v : 0.0f;
                myH[(r + lhalf * 8) * HSTR + col] = v;       // ds_store_b32
            }
        }

        // GEMM2: 16x64 (LDS) vs 64x64 W2
        v8f o[4];
        #pragma unroll
        for (int nt = 0; nt < 4; ++nt) o[nt] = vz;

        #pragma unroll
        for (int kk = 0; kk < 16; ++kk) {
            const int koff = kk * 4 + lhalf * 2;
            const v2f a = *(const v2f*)(aH + koff);          // ds b64
            #pragma unroll
            for (int nt = 0; nt < 4; ++nt) {
                const v2f b = *(const v2f*)(bw2 + kk * KSTR + nt * 32);
                o[nt] = __builtin_amdgcn_wmma_f32_16x16x4_f32(
                    false, a, false, b, (short)0, o[nt], false, false);
            }
        }

        // bias + guarded store (divergence only after the last WMMA of tile)
        #pragma unroll
        for (int r = 0; r < 8; ++r) {
            const int m = row0 + mt * 16 + r + lhalf * 8;
            if (m < N_NODES) {
                #pragma unroll
                for (int nt = 0; nt < 4; ++nt) {
                    const int col = nt * 16 + l16;
                    out[(size_t)m * D + col] = o[nt][r] + sB2[col];
                }
            }
        }
    }
}

// ---------------------------------------------------------------------------
extern "C" void kernel_launch(void* const* d_in, const int* in_sizes, int n_in,
                              void* d_out, int out_size, void* d_ws, size_t ws_size,
                              hipStream_t stream) {
    const float* x  = (const float*)d_in[0];
    const int*   ei = (const int*)d_in[1];          // [2, N_EDGES]
    const float* W1 = (const float*)d_in[2];
    const float* b1 = (const float*)d_in[3];
    const float* W2 = (const float*)d_in[4];
    const float* b2 = (const float*)d_in[5];
    float*       out = (float*)d_out;

    const int* src = ei;
    const int* dst = ei + N_EDGES;

    float* h0 = (float*)d_ws;                       // [N_NODES, D] accumulator

    // 1) h0 = x
    {
        const int n4 = N_NODES * D / 4;
        gin_init<<<(n4 + 255) / 256, 256, 0, stream>>>(x, h0);
    }
    // 2) h0[dst] += x[src]  (16 threads per edge)
    {
        const long long nt = (long long)N_EDGES * 16;
        gin_scatter<<<(int)((nt + 255) / 256), 256, 0, stream>>>(x, src, dst, h0);
    }
    // 3) out = relu(h0@W1+b1)@W2 + b2, fused, f32 WMMA
    {
        const int rows_per_block = MLP_WAVES * 32;  // 128
        gin_mlp<<<(N_NODES + rows_per_block - 1) / rows_per_block, 128, 0, stream>>>(
            h0, W1, b1, W2, b2, out);
    }
}